// InvResMLP_14912126451981
// MI455X (gfx1250) — compile-verified
//
#include <hip/hip_runtime.h>
#include <hip/hip_bf16.h>
#include <stdint.h>

typedef float v2f __attribute__((ext_vector_type(2)));
typedef float v8f __attribute__((ext_vector_type(8)));

#define BATCH 4
#define NPTS 4096
#define NPOINT (BATCH * NPTS)     // 16384
#define DPTS 64
#define NS 32
#define C_IN 67
#define K0PAD 68                   // pad 67 -> 68 (17 k-steps of 4)
#define R2 0.04f
#define BN_EPS 1e-5f

// Toggle for CDNA5 async global->LDS copies (ASYNCcnt-tracked). If the
// assembler rejects the mnemonic, set to 0 and the plain-LDS path is used.
#define USE_ASYNC_LDS 1

// ---------------------------------------------------------------------------
// Async global->LDS b128 copy (GVS addressing: saddr base + 32-bit voffset).
// Per ISA 15.18.3 opcode 98; tracked with ASYNCcnt.
// ---------------------------------------------------------------------------
__device__ __forceinline__ void async_copy_f4(uint32_t lds_addr,
                                              uint32_t goff_bytes,
                                              const void* base) {
#if USE_ASYNC_LDS
    asm volatile("global_load_async_to_lds_b128 %0, %1, %2"
                 :: "v"(lds_addr), "v"(goff_bytes), "s"(base)
                 : "memory");
#endif
}
__device__ __forceinline__ void async_wait0() {
#if USE_ASYNC_LDS
    asm volatile("s_wait_asynccnt 0x0" ::: "memory");
#endif
}
__device__ __forceinline__ uint32_t lds_addr_of(const void* p) {
    return (uint32_t)(uintptr_t)p;   // low 32 bits of flat LDS pointer = LDS offset
}

// ---------------------------------------------------------------------------
// init: zero BN stats accumulators, copy xyz passthrough to output head
// ---------------------------------------------------------------------------
__global__ void init_kernel(const float* __restrict__ xyz,
                            float* __restrict__ out,
                            float* __restrict__ stats) {
    int i = blockIdx.x * 256 + threadIdx.x;
    if (i < 1024) stats[i] = 0.f;
    if (i < NPOINT * 3) out[i] = xyz[i];
}

// ---------------------------------------------------------------------------
// Pack W [O][C] into WMMA B-fragment-major layout:
//   Wfrag[((kt*ntiles + ct)*32 + lane)] = float2{ W[o][k0], W[o][k0+1] }
//   o  = ct*16 + (lane&15),  k0 = kt*4 + (lane>>4)*2   (ISA B layout)
// mode 0: k identity (pad k>=C with 0)
// mode 1: layer-0 channel permutation (points first, rel-xyz last):
//         logical k<64 -> original 3+k ; k in [64,66] -> k-64 ; k>=67 -> pad
// ---------------------------------------------------------------------------
__device__ __forceinline__ int remap_k(int k, int mode, int C) {
    if (mode == 0) return (k < C) ? k : -1;
    if (k < 64) return 3 + k;
    if (k < 67) return k - 64;
    return -1;
}

__global__ void pack_bfrag_kernel(const float* __restrict__ W,
                                  v2f* __restrict__ Wfrag,
                                  int O, int C, int Kt, int mode) {
    int i = blockIdx.x * 256 + threadIdx.x;
    int ntiles = O >> 4;
    int total = Kt * ntiles * 32;
    if (i >= total) return;
    int lane = i & 31;
    int t = i >> 5;
    int kt = t / ntiles;
    int ct = t - kt * ntiles;
    int o  = ct * 16 + (lane & 15);
    int k0 = kt * 4 + ((lane >> 4) << 1);
    int ka = remap_k(k0,     mode, C);
    int kb = remap_k(k0 + 1, mode, C);
    v2f f;
    f.x = (ka >= 0) ? W[o * C + ka] : 0.f;
    f.y = (kb >= 0) ? W[o * C + kb] : 0.f;
    Wfrag[i] = f;
}

// ---------------------------------------------------------------------------
// Ball query: one wave32 per point; streaming scan collects the 32
// lowest-index neighbors within RADIUS (exactly reference semantics).
// ---------------------------------------------------------------------------
__global__ __launch_bounds__(128) void ball_query_kernel(
        const float* __restrict__ xyz, int* __restrict__ gidx) {
    int wave = (blockIdx.x * 128 + threadIdx.x) >> 5;
    int lane = threadIdx.x & 31;
    if (wave >= NPOINT) return;
    int b = wave >> 12;
    const float cx = xyz[wave * 3 + 0];
    const float cy = xyz[wave * 3 + 1];
    const float cz = xyz[wave * 3 + 2];
    const float* base = xyz + (size_t)b * NPTS * 3;
    int cnt = 0;
    int first = -1;
    for (int m0 = 0; m0 < NPTS; m0 += 32) {
        int m = m0 + lane;
        float dx = base[m * 3 + 0] - cx;
        float dy = base[m * 3 + 1] - cy;
        float dz = base[m * 3 + 2] - cz;
        bool within = (dx * dx + dy * dy + dz * dz) <= R2;
        unsigned mask = (unsigned)__ballot(within);   // wave32: low 32 bits
        if (first < 0 && mask) first = m0 + (__ffs(mask) - 1);
        if (within) {
            int pos = cnt + __popc(mask & ((1u << lane) - 1u));
            if (pos < NS) gidx[wave * NS + pos] = m;
        }
        cnt += __popc(mask);
        if (cnt >= NS) break;                          // uniform branch
    }
    if (cnt < NS && lane >= cnt && lane < NS)
        gidx[wave * NS + lane] = first;                // pad with first (lowest-index) neighbor
}

// ---------------------------------------------------------------------------
// Layer 0: fused gather (async global->LDS) + 32x68 @ 68x128 WMMA GEMM +
// bias + neighbor maxpool + per-channel sum/sumsq (BN stats over pre-pool
// activations). One workgroup (4 waves) per point.
// feat channel order: [0..63] point feats, [64..66] rel xyz, [67] zero pad
// (w0 fragments are permuted identically by pack_bfrag_kernel mode 1).
// ---------------------------------------------------------------------------
__global__ __launch_bounds__(128) void layer0_kernel(
        const float* __restrict__ xyz, const float* __restrict__ points,
        const int* __restrict__ gidx, const v2f* __restrict__ w0frag,
        const float* __restrict__ bias0, float* __restrict__ ymax,
        float* __restrict__ stats0) {
    __shared__ __align__(16) float feat[NS][K0PAD];    // row stride 272B (16B-aligned)
    int p = blockIdx.x;                                // (b,n) flattened
    int b = p >> 12;
    int tid = threadIdx.x;

    // gather 32 x 64 point features: 4 threads per neighbor row, b128 chunks
    {
        int s  = tid >> 2;
        int c4 = tid & 3;
        int idx = gidx[p * NS + s];
        uint32_t goff = (uint32_t)(((b * NPTS + idx) * DPTS) * 4);
        uint32_t lrow = lds_addr_of(&feat[s][0]);
#if USE_ASYNC_LDS
        for (int q = c4; q < 16; q += 4)
            async_copy_f4(lrow + q * 16, goff + q * 16, points);
#else
        for (int q = c4; q < 16; q += 4)
            *(float4*)&feat[s][q * 4] =
                *(const float4*)((const char*)points + goff + q * 16);
#endif
    }
    // rel-xyz tail channels: one thread per neighbor row
    if (tid < NS) {
        const float cx = xyz[p * 3 + 0];
        const float cy = xyz[p * 3 + 1];
        const float cz = xyz[p * 3 + 2];
        int idx = gidx[p * NS + tid];
        const float* xs = xyz + (size_t)(b * NPTS + idx) * 3;
        feat[tid][64] = xs[0] - cx;
        feat[tid][65] = xs[1] - cy;
        feat[tid][66] = xs[2] - cz;
        feat[tid][67] = 0.f;
    }
    async_wait0();
    __syncthreads();

    int wv = tid >> 5, lane = tid & 31;
    int lm = lane & 15;
    int hk = (lane >> 4) << 1;                          // half-wave K select
    for (int cti = 0; cti < 2; ++cti) {
        int ct = wv + cti * 4;                          // column tile 0..7
        float bv  = bias0[ct * 16 + lm];
        float bmx = -3.402823466e38f, ssum = 0.f, ssq = 0.f;
        for (int rt = 0; rt < 2; ++rt) {
            v8f acc = {0.f, 0.f, 0.f, 0.f, 0.f, 0.f, 0.f, 0.f};
            int row = rt * 16 + lm;
            for (int kt = 0; kt < K0PAD / 4; ++kt) {
                int kk = kt * 4 + hk;
                v2f a  = { feat[row][kk], feat[row][kk + 1] };
                v2f bf = w0frag[(kt * 8 + ct) * 32 + lane];   // one b64 load
                acc = __builtin_amdgcn_wmma_f32_16x16x4_f32(
                          false, a, false, bf, (short)0, acc, false, false);
            }
#pragma unroll
            for (int j = 0; j < 8; ++j) {
                float y = acc[j] + bv;
                bmx = fmaxf(bmx, y);
                ssum += y;
                ssq  += y * y;
            }
        }
        // combine the two lane halves (same channel, rows M vs M+8)
        bmx  = fmaxf(bmx, __shfl_xor(bmx, 16));
        ssum += __shfl_xor(ssum, 16);
        ssq  += __shfl_xor(ssq, 16);
        if (lane < 16) {
            int o = ct * 16 + lane;
            ymax[(size_t)p * 128 + o] = bmx;            // pre-BN maxpooled
            atomicAdd(&stats0[o],       ssum);
            atomicAdd(&stats0[128 + o], ssq);
        }
    }
}

// ---------------------------------------------------------------------------
// Layers 1/2: [16384 x 128] @ [128 x Ncols] WMMA GEMM + bias + BN stats.
// 256-thread block computes a 32x64 tile. A staged in LDS (async copy);
// B fragments read directly from the packed fragment buffer (L2 resident).
// ---------------------------------------------------------------------------
__global__ __launch_bounds__(256) void gemm_bias_stats_kernel(
        const float* __restrict__ X, const v2f* __restrict__ Bfrag,
        const float* __restrict__ bias, float* __restrict__ Y,
        float* __restrict__ stats, int Ncols, int ntiles) {
    __shared__ __align__(16) float As[32][132];         // row stride 528B (16B-aligned)
    int tid  = threadIdx.x;
    int row0 = blockIdx.x * 32;
    int col0 = blockIdx.y * 64;
    // stage 32 x 128 A tile
    for (int i = tid; i < 32 * 32; i += 256) {          // float4 units
        int r = i >> 5, c4 = i & 31;
        uint32_t lds = lds_addr_of(&As[r][c4 * 4]);
        uint32_t goff = (uint32_t)((((row0 + r) * 128) + c4 * 4) * 4);
#if USE_ASYNC_LDS
        async_copy_f4(lds, goff, X);
#else
        *(float4*)&As[r][c4 * 4] = *(const float4*)((const char*)X + goff);
#endif
    }
    async_wait0();
    __syncthreads();

    int wv = tid >> 5, lane = tid & 31;
    int lm = lane & 15;
    int hk = (lane >> 4) << 1;
    int rt = wv & 1;                                    // 2 row tiles
    int ct = wv >> 1;                                   // 4 col tiles
    int ctg = (col0 >> 4) + ct;                         // global column tile
    v8f acc = {0.f, 0.f, 0.f, 0.f, 0.f, 0.f, 0.f, 0.f};
    int row = rt * 16 + lm;
    for (int kt = 0; kt < 32; ++kt) {
        int kk = kt * 4 + hk;
        v2f a  = { As[row][kk], As[row][kk + 1] };      // ds_load_b64
        v2f bf = Bfrag[(kt * ntiles + ctg) * 32 + lane];// global b64
        acc = __builtin_amdgcn_wmma_f32_16x16x4_f32(
                  false, a, false, bf, (short)0, acc, false, false);
    }
    int o = col0 + ct * 16 + lm;
    float bv = bias[o];
    float ssum = 0.f, ssq = 0.f;
    int rbase = row0 + rt * 16 + ((lane >> 4) << 3);    // M=j or j+8 per half
#pragma unroll
    for (int j = 0; j < 8; ++j) {
        float y = acc[j] + bv;
        Y[(size_t)(rbase + j) * Ncols + o] = y;
        ssum += y;
        ssq  += y * y;
    }
    atomicAdd(&stats[o],         ssum);
    atomicAdd(&stats[Ncols + o], ssq);
}

// ---------------------------------------------------------------------------
// BN (training-mode batch stats) + ReLU, elementwise. invcnt = 1/#elements
// the stats were accumulated over (layer0: B*ns*N; layers1/2: B*N).
// ---------------------------------------------------------------------------
__global__ void bn_relu_kernel(const float* __restrict__ Yin,
                               float* __restrict__ Xout,
                               const float* __restrict__ stats,
                               const float* __restrict__ gamma,
                               const float* __restrict__ beta,
                               float invcnt, int cmask, int ncols, int total) {
    int i = blockIdx.x * 256 + threadIdx.x;
    if (i >= total) return;
    int o = i & cmask;
    float mean = stats[o] * invcnt;
    float var  = stats[ncols + o] * invcnt - mean * mean;
    float y = Yin[i];
    float x = (y - mean) * rsqrtf(var + BN_EPS) * gamma[o] + beta[o];
    Xout[i] = fmaxf(x, 0.f);
}

// ---------------------------------------------------------------------------
extern "C" void kernel_launch(void* const* d_in, const int* in_sizes, int n_in,
                              void* d_out, int out_size, void* d_ws, size_t ws_size,
                              hipStream_t stream) {
    (void)in_sizes; (void)n_in; (void)out_size; (void)ws_size;
    const float* xyz    = (const float*)d_in[0];
    const float* points = (const float*)d_in[1];
    const float* w0  = (const float*)d_in[2];
    const float* b0  = (const float*)d_in[3];
    const float* g0  = (const float*)d_in[4];
    const float* bt0 = (const float*)d_in[5];
    const float* w1  = (const float*)d_in[6];
    const float* b1  = (const float*)d_in[7];
    const float* g1  = (const float*)d_in[8];
    const float* bt1 = (const float*)d_in[9];
    const float* w2  = (const float*)d_in[10];
    const float* b2  = (const float*)d_in[11];
    const float* g2  = (const float*)d_in[12];
    const float* bt2 = (const float*)d_in[13];
    float* out = (float*)d_out;

    // workspace layout (floats)
    float* ws    = (float*)d_ws;
    float* w0f   = ws;                      // 17*8*32*2  = 8704
    float* w1f   = w0f + 8704;              // 32*8*32*2  = 16384
    float* w2f   = w1f + 16384;             // 32*16*32*2 = 32768
    float* stats = w2f + 32768;             // 1024 (s0:256, s1:256, s2:512)
    int*   gidx  = (int*)(stats + 1024);    // 16384*32 ints
    float* bufA  = (float*)(gidx + NPOINT * NS);  // 16384*128
    float* bufB  = bufA + (size_t)NPOINT * 128;   // 16384*128
    float* bufC  = bufB + (size_t)NPOINT * 128;   // 16384*256

    // 1) zero stats, copy xyz to output head
    init_kernel<<<(NPOINT * 3 + 255) / 256, 256, 0, stream>>>(xyz, out, stats);
    // 2) pack weights into WMMA B-fragment layout (w0 with channel permute)
    pack_bfrag_kernel<<<(17 * 8 * 32 + 255) / 256, 256, 0, stream>>>(
        w0, (v2f*)w0f, 128, 67, 17, 1);
    pack_bfrag_kernel<<<(32 * 8 * 32 + 255) / 256, 256, 0, stream>>>(
        w1, (v2f*)w1f, 128, 128, 32, 0);
    pack_bfrag_kernel<<<(32 * 16 * 32 + 255) / 256, 256, 0, stream>>>(
        w2, (v2f*)w2f, 256, 128, 32, 0);
    // 3) ball query
    ball_query_kernel<<<NPOINT / 4, 128, 0, stream>>>(xyz, gidx);
    // 4) layer 0: gather + GEMM + maxpool + stats
    layer0_kernel<<<NPOINT, 128, 0, stream>>>(
        xyz, points, gidx, (const v2f*)w0f, b0, bufA, stats);
    // 5) BN0 + ReLU on pooled activations (stats over B*ns*N = 524288)
    bn_relu_kernel<<<(NPOINT * 128 + 255) / 256, 256, 0, stream>>>(
        bufA, bufA, stats, g0, bt0, 1.f / 524288.f, 127, 128, NPOINT * 128);
    // 6) layer 1 GEMM + stats
    gemm_bias_stats_kernel<<<dim3(NPOINT / 32, 2), 256, 0, stream>>>(
        bufA, (const v2f*)w1f, b1, bufB, stats + 256, 128, 8);
    bn_relu_kernel<<<(NPOINT * 128 + 255) / 256, 256, 0, stream>>>(
        bufB, bufB, stats + 256, g1, bt1, 1.f / 16384.f, 127, 128, NPOINT * 128);
    // 7) layer 2 GEMM + stats, BN straight into output tail
    gemm_bias_stats_kernel<<<dim3(NPOINT / 32, 4), 256, 0, stream>>>(
        bufB, (const v2f*)w2f, b2, bufC, stats + 512, 256, 16);
    bn_relu_kernel<<<(NPOINT * 256 + 255) / 256, 256, 0, stream>>>(
        bufC, out + NPOINT * 3, stats + 512, g2, bt2, 1.f / 16384.f, 255, 256, NPOINT * 256);
}